// GCN_12661563588776
// MI455X (gfx1250) — compile-verified
//
#include <hip/hip_runtime.h>
#include <math.h>

// ---------------------------------------------------------------------------
// GraphConv GNN inference for gfx1250 (MI455X), wave32 + V_WMMA_F32_16X16X4_F32
//
// Restructure: agg(x[src]) @ Wr == agg((x@Wr)[src])  (linearity), so project
// to D_HID=32 first, then scatter-add 32-dim messages (6.25x less traffic
// than the reference order for layer 1).
// ---------------------------------------------------------------------------

typedef __attribute__((ext_vector_type(2))) float v2f;
typedef __attribute__((ext_vector_type(8))) float v8f;

#define D_HID 32
#define MAX_K 200
#define N_GRAPHS 64

// R[m,:] = X[m,:] @ Wr              (message projection, fed to edge scatter)
// S[m,:] = X[m,:] @ Wl + b          (root term, becomes the accumulator)
// X: [M,K] row-major (K multiple of 4), Wr/Wl: [K,32] row-major.
// 256 threads = 8 waves; each wave computes a 16x32 strip via WMMA f32 16x16x4.
__global__ __launch_bounds__(256) void gemm_dual_wmma(
    const float* __restrict__ X, const float* __restrict__ Wr,
    const float* __restrict__ Wl, const float* __restrict__ b,
    float* __restrict__ R, float* __restrict__ S, int M, int K)
{
  __shared__ float ldsW[2 * MAX_K * D_HID];   // 51.2 KB of the 320 KB WGP LDS
  const int tid = threadIdx.x;
  for (int i = tid; i < K * D_HID; i += 256) {
    ldsW[i]                   = Wr[i];
    ldsW[MAX_K * D_HID + i]   = Wl[i];
  }
  __syncthreads();

  const int wave = tid >> 5;
  const int lane = tid & 31;
  const int half = lane >> 4;     // 0: lanes 0-15, 1: lanes 16-31
  const int lml  = lane & 15;

  const int mbase = (blockIdx.x * 8 + wave) * 16;
  if (mbase >= M) return;         // wave-uniform exit (EXEC stays all-1s)

  // A-matrix 16x4 fp32 layout: lanes 0-15 -> {K=0,K=1}, lanes 16-31 -> {K=2,K=3}
  int arow = mbase + lml;
  if (arow >= M) arow = M - 1;    // clamp: padded rows compute garbage, never stored
  const float* __restrict__ Xrow = X + (long)arow * K;

  v8f cr0 = {}, cr1 = {}, cl0 = {}, cl1 = {};
  for (int k0 = 0; k0 < K; k0 += 4) {
    const int ka = k0 + 2 * half;
    v2f a;
    a.x = Xrow[ka];
    a.y = Xrow[ka + 1];

    // B-matrix 4x16 fp32: row-striped across lanes; vgpr v -> K = ka+v, N = n0+lml
    const float* wr = ldsW + ka * D_HID;
    const float* wl = ldsW + MAX_K * D_HID + ka * D_HID;
    v2f br0, br1, bl0, bl1;
    br0.x = wr[lml];          br0.y = wr[D_HID + lml];
    br1.x = wr[16 + lml];     br1.y = wr[D_HID + 16 + lml];
    bl0.x = wl[lml];          bl0.y = wl[D_HID + lml];
    bl1.x = wl[16 + lml];     bl1.y = wl[D_HID + 16 + lml];

    cr0 = __builtin_amdgcn_wmma_f32_16x16x4_f32(false, a, false, br0, (short)0, cr0, false, false);
    cr1 = __builtin_amdgcn_wmma_f32_16x16x4_f32(false, a, false, br1, (short)0, cr1, false, false);
    cl0 = __builtin_amdgcn_wmma_f32_16x16x4_f32(false, a, false, bl0, (short)0, cl0, false, false);
    cl1 = __builtin_amdgcn_wmma_f32_16x16x4_f32(false, a, false, bl1, (short)0, cl1, false, false);
  }

  // C/D layout: VGPR r -> lanes 0-15: M=r, lanes 16-31: M=r+8; N = lane%16 (+n0)
  const float bia0 = b[lml];
  const float bia1 = b[16 + lml];
  for (int r = 0; r < 8; ++r) {
    const int row = mbase + r + 8 * half;
    if (row < M) {
      R[(long)row * D_HID + lml]      = cr0[r];
      R[(long)row * D_HID + 16 + lml] = cr1[r];
      S[(long)row * D_HID + lml]      = cl0[r] + bia0;
      S[(long)row * D_HID + 16 + lml] = cl1[r] + bia1;
    }
  }
}

// S[dst,:] += R[src,:] over all edges; one thread per (edge, feature).
__global__ __launch_bounds__(256) void edge_scatter_add(
    const int* __restrict__ src, const int* __restrict__ dst,
    const float* __restrict__ R, float* __restrict__ S, int n_edges)
{
  const int gid = blockIdx.x * 256 + threadIdx.x;
  const int e = gid >> 5;
  const int f = gid & 31;
  if (e >= n_edges) return;
  const int s = src[e];
  const int d = dst[e];
  atomicAdd(&S[(long)d * D_HID + f], R[(long)s * D_HID + f]);
}

// In-place ELU (alpha = 1)
__global__ __launch_bounds__(256) void elu_inplace(float* __restrict__ H, int n)
{
  const int i = blockIdx.x * 256 + threadIdx.x;
  if (i < n) {
    const float v = H[i];
    H[i] = v > 0.0f ? v : (expf(v) - 1.0f);
  }
}

__global__ __launch_bounds__(256) void zero_buf(float* __restrict__ p, int n)
{
  const int i = blockIdx.x * 256 + threadIdx.x;
  if (i < n) p[i] = 0.0f;
}

// Per-graph sums + counts
__global__ __launch_bounds__(256) void pool_accum(
    const float* __restrict__ H, const int* __restrict__ batch,
    float* __restrict__ sums, float* __restrict__ cnt, int M)
{
  const int gid = blockIdx.x * 256 + threadIdx.x;
  const int nidx = gid >> 5;
  const int f = gid & 31;
  if (nidx >= M) return;
  const int g = batch[nidx];
  atomicAdd(&sums[g * D_HID + f], H[(long)nidx * D_HID + f]);
  if (f == 0) atomicAdd(&cnt[g], 1.0f);
}

// pooled @ Wlin + blin, log_softmax over 2 classes. One thread per graph.
__global__ __launch_bounds__(64) void head_kernel(
    const float* __restrict__ sums, const float* __restrict__ cnt,
    const float* __restrict__ Wlin, const float* __restrict__ blin,
    float* __restrict__ out)
{
  const int g = threadIdx.x;
  if (g >= N_GRAPHS) return;
  const float inv = 1.0f / fmaxf(cnt[g], 1.0f);
  float l0 = blin[0], l1 = blin[1];
  for (int k = 0; k < D_HID; ++k) {
    const float p = sums[g * D_HID + k] * inv;
    l0 += p * Wlin[k * 2 + 0];
    l1 += p * Wlin[k * 2 + 1];
  }
  const float m = fmaxf(l0, l1);
  const float lse = m + logf(expf(l0 - m) + expf(l1 - m));
  out[g * 2 + 0] = l0 - lse;
  out[g * 2 + 1] = l1 - lse;
}

extern "C" void kernel_launch(void* const* d_in, const int* in_sizes, int n_in,
                              void* d_out, int out_size, void* d_ws, size_t ws_size,
                              hipStream_t stream) {
  const float* x     = (const float*)d_in[0];   // [M, 200]
  const int*   ei    = (const int*)d_in[1];     // [2, E]
  // d_in[2] = edge_attr (unused by reference GraphConv)
  const int*   batch = (const int*)d_in[3];     // [M]
  const float* W1r = (const float*)d_in[4];
  const float* W1l = (const float*)d_in[5];
  const float* b1  = (const float*)d_in[6];
  const float* W2r = (const float*)d_in[7];
  const float* W2l = (const float*)d_in[8];
  const float* b2  = (const float*)d_in[9];
  const float* W3r = (const float*)d_in[10];
  const float* W3l = (const float*)d_in[11];
  const float* b3  = (const float*)d_in[12];
  const float* Wlin = (const float*)d_in[13];
  const float* blin = (const float*)d_in[14];
  float* out = (float*)d_out;

  const int M  = in_sizes[0] / 200;   // 50000 nodes
  const int E  = in_sizes[1] / 2;     // 400000 edges
  const int K1 = 200;

  const int* src = ei;
  const int* dst = ei + E;

  // Workspace layout (floats): bufA (messages R), bufB (root/accum S -> h), pool
  float* ws   = (float*)d_ws;
  float* bufA = ws;                          // [M*32]
  float* bufB = ws + (long)M * D_HID;        // [M*32]
  float* sums = ws + 2L * M * D_HID;         // [64*32]
  float* cnt  = sums + N_GRAPHS * D_HID;     // [64]

  const int gemmGrid = (M + 127) / 128;                   // 8 waves x 16 rows per block
  const int edgeGrid = (E * 32 + 255) / 256;
  const int eluGrid  = (M * D_HID + 255) / 256;
  const int poolN    = N_GRAPHS * D_HID + N_GRAPHS;

  // ---- Layer 1 (K=200) ----
  gemm_dual_wmma<<<gemmGrid, 256, 0, stream>>>(x, W1r, W1l, b1, bufA, bufB, M, K1);
  edge_scatter_add<<<edgeGrid, 256, 0, stream>>>(src, dst, bufA, bufB, E);
  elu_inplace<<<eluGrid, 256, 0, stream>>>(bufB, M * D_HID);

  // ---- Layer 2 (K=32) ----  (reads bufB rows then overwrites same rows: safe per-block)
  gemm_dual_wmma<<<gemmGrid, 256, 0, stream>>>(bufB, W2r, W2l, b2, bufA, bufB, M, D_HID);
  edge_scatter_add<<<edgeGrid, 256, 0, stream>>>(src, dst, bufA, bufB, E);
  elu_inplace<<<eluGrid, 256, 0, stream>>>(bufB, M * D_HID);

  // ---- Layer 3 (K=32) ----
  gemm_dual_wmma<<<gemmGrid, 256, 0, stream>>>(bufB, W3r, W3l, b3, bufA, bufB, M, D_HID);
  edge_scatter_add<<<edgeGrid, 256, 0, stream>>>(src, dst, bufA, bufB, E);
  elu_inplace<<<eluGrid, 256, 0, stream>>>(bufB, M * D_HID);

  // ---- Global mean pool + classifier head ----
  zero_buf<<<(poolN + 255) / 256, 256, 0, stream>>>(sums, poolN);
  pool_accum<<<(M * 32 + 255) / 256, 256, 0, stream>>>(bufB, batch, sums, cnt, M);
  head_kernel<<<1, 64, 0, stream>>>(sums, cnt, Wlin, blin, out);
}